// Simplicial_GCN_73924977099259
// MI455X (gfx1250) — compile-verified
//
#include <hip/hip_runtime.h>
#include <math.h>

typedef __attribute__((ext_vector_type(2))) float v2f;
typedef __attribute__((ext_vector_type(8))) float v8f;

// ---- order-preserving float<->uint map for atomic float max (handles negatives) ----
__device__ __forceinline__ unsigned f2ord(float f) {
    unsigned u = __float_as_uint(f);
    return (u & 0x80000000u) ? ~u : (u | 0x80000000u);
}
__device__ __forceinline__ float ord2f(unsigned u) {
    return (u & 0x80000000u) ? __uint_as_float(u & 0x7fffffffu)
                             : __uint_as_float(~u);
}

// ============================================================================
// Head-aware pad/copy: src is [HK*CK x HN*CN] row-major, dst is [HK*CKP x HN*CNP]
// with element (hk,ck,hn,cn) -> dst[(hk*CKP+ck)*HN*CNP + hn*CNP+cn]; pads = 0.
// ============================================================================
__global__ void gat_pad_w(const float* __restrict__ src, float* __restrict__ dst,
                          int HK, int CK, int CKP, int HN, int CN, int CNP) {
    int Np = HN * CNP;
    int tot = HK * CKP * Np;
    int t = blockIdx.x * blockDim.x + threadIdx.x;
    if (t >= tot) return;
    int kp = t / Np, np = t - kp * Np;
    int hk = kp / CKP, ck = kp - hk * CKP;
    int hn = np / CNP, cn = np - hn * CNP;
    float v = 0.0f;
    if (ck < CK && cn < CN)
        v = src[(hk * CK + ck) * (HN * CN) + hn * CN + cn];
    dst[t] = v;
}

// ============================================================================
// Bounds-check-free, software-pipelined fp32 GEMM via V_WMMA_F32_16X16X4_F32.
// Requires: M % 16 == 0, K % 4 == 0 and K >= 4, NC = NT*16.
// A frag (16x4 f32): lanes 0-15 M=lane K={0,1}; lanes 16-31 M=lane-16 K={2,3}.
// B frag (4x16):     lanes 0-15 K={0,1}; lanes 16-31 K={2,3}; N=col0+(lane&15).
// C/D: VGPR i -> row i (lanes 0-15) / row i+8 (lanes 16-31).
// ============================================================================
template <int NT>
__global__ void gat_gemm_wmma(const float* __restrict__ A, int lda,
                              const float* __restrict__ W, int ldw,
                              const float* __restrict__ bias,
                              float* __restrict__ O, int ldo,
                              int M, int K) {
    int wave = blockIdx.x * (blockDim.x >> 5) + (threadIdx.x >> 5);
    int lane = threadIdx.x & 31;
    if (wave >= (M >> 4)) return;                  // uniform per wave
    int row0 = wave << 4;
    int half = lane >> 4;                          // K sub-pair select
    int l    = lane & 15;

    v8f c[NT];
#pragma unroll
    for (int t = 0; t < NT; ++t) c[t] = (v8f){};

    const float* ap = A + (size_t)(row0 + l) * lda + (half << 1);
    const float* bp = W + (size_t)(half << 1) * ldw + l;
    const size_t bstep = (size_t)4 * ldw;

    // prologue: fragment for k0 = 0
    v2f a = *(const v2f*)ap;
    v2f b[NT];
#pragma unroll
    for (int t = 0; t < NT; ++t) { b[t].x = bp[t * 16]; b[t].y = bp[t * 16 + ldw]; }

    for (int k0 = 4; k0 < K; k0 += 4) {
        ap += 4;
        bp += bstep;
        // prefetch next fragment before consuming current one
        v2f an = *(const v2f*)ap;
        v2f bn[NT];
#pragma unroll
        for (int t = 0; t < NT; ++t) { bn[t].x = bp[t * 16]; bn[t].y = bp[t * 16 + ldw]; }
#pragma unroll
        for (int t = 0; t < NT; ++t)
            c[t] = __builtin_amdgcn_wmma_f32_16x16x4_f32(
                       false, a, false, b[t], (short)0, c[t], false, false);
        a = an;
#pragma unroll
        for (int t = 0; t < NT; ++t) b[t] = bn[t];
    }
#pragma unroll
    for (int t = 0; t < NT; ++t)
        c[t] = __builtin_amdgcn_wmma_f32_16x16x4_f32(
                   false, a, false, b[t], (short)0, c[t], false, false);

#pragma unroll
    for (int t = 0; t < NT; ++t) {
        int oc = t * 16 + l;
        float bv = bias ? bias[oc] : 0.0f;
        float* op = O + (size_t)(row0 + (half << 3)) * ldo + oc;
#pragma unroll
        for (int i = 0; i < 8; ++i)
            op[(size_t)i * ldo] = c[t][i] + bv;
    }
}

// ---- init emax (-inf ordered), denom=0, aggregation buffer (padded) = 0 ----
__global__ void gat_init(unsigned* __restrict__ emax, float* __restrict__ denom,
                         float* __restrict__ agg, int NH, int NPAD) {
    int t = blockIdx.x * blockDim.x + threadIdx.x;
    if (t < NH) { emax[t] = f2ord(-INFINITY); denom[t] = 0.0f; }
    if (t < NPAD) agg[t] = 0.0f;
}

// ---- pass 1: e = <leaky_relu(xl[src]+xr[dst]), att[h]>, atomic segment-max.
//      Head-padded layout: per-head block of 32 floats (pads are exact zeros),
//      so all gathers are aligned float4 (global_load_b128). ----
__global__ void gat_edge_score(const int* __restrict__ ei, int E, int N, int H,
                               int hcPad,
                               const float* __restrict__ xl, const float* __restrict__ xr,
                               const float* __restrict__ attp,   // [H x 32], zero-padded
                               float* __restrict__ ew, unsigned* __restrict__ emax) {
    long long t = (long long)blockIdx.x * blockDim.x + threadIdx.x;
    if (t >= (long long)(E + N) * H) return;
    int idx = (int)(t / H);
    int h   = (int)(t - (long long)idx * H);
    int src, dst;
    if (idx < E) { src = ei[idx]; dst = ei[E + idx]; }
    else         { src = dst = idx - E; }                    // self-loop
    const float4* pl = (const float4*)(xl + (long long)src * hcPad + h * 32);
    const float4* pr = (const float4*)(xr + (long long)dst * hcPad + h * 32);
    const float4* pa = (const float4*)(attp + h * 32);
    float e = 0.0f;
#pragma unroll
    for (int q = 0; q < 8; ++q) {                            // 32 floats (2 pad = 0)
        float4 L = pl[q], R = pr[q], A4 = pa[q];
        float m;
        m = L.x + R.x; m = (m > 0.0f) ? m : 0.2f * m; e += m * A4.x;
        m = L.y + R.y; m = (m > 0.0f) ? m : 0.2f * m; e += m * A4.y;
        m = L.z + R.z; m = (m > 0.0f) ? m : 0.2f * m; e += m * A4.z;
        m = L.w + R.w; m = (m > 0.0f) ? m : 0.2f * m; e += m * A4.w;
    }
    ew[t] = e;
    atomicMax(&emax[(long long)dst * H + h], f2ord(e));
}

// ---- pass 2: ex = exp(e - emax[dst]); atomic segment-sum ----
__global__ void gat_edge_exp(const int* __restrict__ ei, int E, int N, int H,
                             const unsigned* __restrict__ emax,
                             float* __restrict__ ew, float* __restrict__ denom) {
    long long t = (long long)blockIdx.x * blockDim.x + threadIdx.x;
    if (t >= (long long)(E + N) * H) return;
    int idx = (int)(t / H);
    int h   = (int)(t - (long long)idx * H);
    int dst = (idx < E) ? ei[E + idx] : (idx - E);
    float ex = expf(ew[t] - ord2f(emax[(long long)dst * H + h]));
    ew[t] = ex;
    atomicAdd(&denom[(long long)dst * H + h], ex);
}

// ---- pass 3: agg[dst] += (ex / denom[dst]) * xl[src]  (real 30 channels) ----
__global__ void gat_edge_aggr(const int* __restrict__ ei, int E, int N, int H, int C,
                              int hcPad,
                              const float* __restrict__ xl, const float* __restrict__ ew,
                              const float* __restrict__ denom, float* __restrict__ agg) {
    long long t = (long long)blockIdx.x * blockDim.x + threadIdx.x;
    if (t >= (long long)(E + N) * H) return;
    int idx = (int)(t / H);
    int h   = (int)(t - (long long)idx * H);
    int src, dst;
    if (idx < E) { src = ei[idx]; dst = ei[E + idx]; }
    else         { src = dst = idx - E; }
    float alpha = ew[t] / denom[(long long)dst * H + h];
    const float* pl = xl  + (long long)src * hcPad + h * 32;
    float*       po = agg + (long long)dst * hcPad + h * 32;
    for (int cc = 0; cc < C; ++cc)
        atomicAdd(&po[cc], alpha * pl[cc]);
}

// ---- bias (+ optional relu) over full padded buffer; biasp pads are zero,
//      and relu(0)=0, so pad columns stay exactly zero. ----
__global__ void gat_bias_act(float* __restrict__ buf, const float* __restrict__ biasp,
                             long long totalPad, int hcPad, int relu) {
    long long t = (long long)blockIdx.x * blockDim.x + threadIdx.x;
    if (t >= totalPad) return;
    float v = buf[t] + biasp[(int)(t % hcPad)];
    if (relu) v = fmaxf(v, 0.0f);
    buf[t] = v;
}

extern "C" void kernel_launch(void* const* d_in, const int* in_sizes, int n_in,
                              void* d_out, int out_size, void* d_ws, size_t ws_size,
                              hipStream_t stream) {
    const int*   ei   = (const int*)  d_in[0];   // [2, E]
    const float* x    = (const float*)d_in[1];   // [N, 128]
    const float* W1l  = (const float*)d_in[2];   // [128, 60]
    const float* W1r  = (const float*)d_in[3];
    const float* att1 = (const float*)d_in[4];   // [2, 30]
    const float* b1   = (const float*)d_in[5];   // [60]
    const float* W2l  = (const float*)d_in[6];   // [60, 30]
    const float* W2r  = (const float*)d_in[7];
    const float* att2 = (const float*)d_in[8];   // [1, 30]
    const float* b2   = (const float*)d_in[9];   // [30]
    const float* W3l  = (const float*)d_in[10];  // [30, 60]
    const float* W3r  = (const float*)d_in[11];
    const float* att3 = (const float*)d_in[12];  // [2, 30]
    const float* b3   = (const float*)d_in[13];  // [60]
    const float* Wlin = (const float*)d_in[14];  // [60, 16]
    const float* blin = (const float*)d_in[15];  // [16]

    const int E = in_sizes[0] / 2;
    const int N = in_sizes[1] / 128;             // 100000, N % 16 == 0

    // ---- workspace carve-up (all offsets multiples of 256B) ----
    char* ws = (char*)d_ws;
    size_t cap64 = (size_t)N * 64 * sizeof(float);
    float*    xl    = (float*)ws;  ws += cap64;                 // per-head stride 32
    float*    xr    = (float*)ws;  ws += cap64;
    float*    hA    = (float*)ws;  ws += cap64;                 // row stride 64
    float*    hB    = (float*)ws;  ws += (size_t)N * 32 * sizeof(float); // row stride 32
    float*    ewk   = (float*)ws;  ws += (size_t)(E + N) * 2 * sizeof(float);
    unsigned* emax  = (unsigned*)ws; ws += (size_t)N * 2 * sizeof(unsigned);
    float*    denom = (float*)ws;  ws += (size_t)N * 2 * sizeof(float);
    float*    W1lp  = (float*)ws;  ws += 128 * 64 * sizeof(float);
    float*    W1rp  = (float*)ws;  ws += 128 * 64 * sizeof(float);
    float*    W2lp  = (float*)ws;  ws += 64 * 32 * sizeof(float);
    float*    W2rp  = (float*)ws;  ws += 64 * 32 * sizeof(float);
    float*    W3lp  = (float*)ws;  ws += 32 * 64 * sizeof(float);
    float*    W3rp  = (float*)ws;  ws += 32 * 64 * sizeof(float);
    float*    Wlp   = (float*)ws;  ws += 64 * 16 * sizeof(float);
    float*    a1p   = (float*)ws;  ws += 64 * sizeof(float);    // att pads
    float*    a2p   = (float*)ws;  ws += 64 * sizeof(float);
    float*    a3p   = (float*)ws;  ws += 64 * sizeof(float);
    float*    b1p   = (float*)ws;  ws += 64 * sizeof(float);    // bias pads
    float*    b2p   = (float*)ws;  ws += 64 * sizeof(float);
    float*    b3p   = (float*)ws;  ws += 64 * sizeof(float);

    auto padw = [&](const float* s, float* d, int HK, int CK, int CKP,
                    int HN, int CN, int CNP) {
        int tot = HK * CKP * HN * CNP;
        gat_pad_w<<<(tot + 255) / 256, 256, 0, stream>>>(s, d, HK, CK, CKP, HN, CN, CNP);
    };
    // weights: K-side / N-side head-aware padding (pads zero)
    padw(W1l, W1lp, 1, 128, 128, 2, 30, 32);   // 128x60 -> 128x64
    padw(W1r, W1rp, 1, 128, 128, 2, 30, 32);
    padw(W2l, W2lp, 2, 30, 32, 1, 30, 32);     // 60x30  -> 64x32
    padw(W2r, W2rp, 2, 30, 32, 1, 30, 32);
    padw(W3l, W3lp, 1, 30, 32, 2, 30, 32);     // 30x60  -> 32x64
    padw(W3r, W3rp, 1, 30, 32, 2, 30, 32);
    padw(Wlin, Wlp, 2, 30, 32, 1, 16, 16);     // 60x16  -> 64x16
    // attention vectors [H x 30] -> [H x 32]; biases likewise
    padw(att1, a1p, 1, 2, 2, 1, 30, 32);
    padw(att2, a2p, 1, 1, 1, 1, 30, 32);
    padw(att3, a3p, 1, 2, 2, 1, 30, 32);
    padw(b1, b1p, 1, 1, 1, 2, 30, 32);
    padw(b2, b2p, 1, 1, 1, 1, 30, 32);
    padw(b3, b3p, 1, 1, 1, 2, 30, 32);

    auto gemm = [&](const float* A, int lda, const float* W, int ldw,
                    const float* bias, float* O, int ldo, int M, int K, int NC) {
        const int wpb = 4;
        int waves = M >> 4;
        int blocks = (waves + wpb - 1) / wpb;
        if (NC == 64)
            gat_gemm_wmma<4><<<blocks, wpb * 32, 0, stream>>>(A, lda, W, ldw, bias, O, ldo, M, K);
        else if (NC == 32)
            gat_gemm_wmma<2><<<blocks, wpb * 32, 0, stream>>>(A, lda, W, ldw, bias, O, ldo, M, K);
        else
            gat_gemm_wmma<1><<<blocks, wpb * 32, 0, stream>>>(A, lda, W, ldw, bias, O, ldo, M, K);
    };

    auto layer = [&](const float* hin, int lda, int Kin,
                     const float* Wl_, const float* Wr_, int hcPad,
                     const float* attp, const float* biasp, int H,
                     float* hout, int relu) {
        gemm(hin, lda, Wl_, hcPad, nullptr, xl, hcPad, N, Kin, hcPad);
        gemm(hin, lda, Wr_, hcPad, nullptr, xr, hcPad, N, Kin, hcPad);
        int NH = N * H, NPAD = N * hcPad;
        gat_init<<<(NPAD + 255) / 256, 256, 0, stream>>>(emax, denom, hout, NH, NPAD);
        long long tot = (long long)(E + N) * H;
        int eb = (int)((tot + 255) / 256);
        gat_edge_score<<<eb, 256, 0, stream>>>(ei, E, N, H, hcPad, xl, xr, attp, ewk, emax);
        gat_edge_exp  <<<eb, 256, 0, stream>>>(ei, E, N, H, emax, ewk, denom);
        gat_edge_aggr <<<eb, 256, 0, stream>>>(ei, E, N, H, 30, hcPad, xl, ewk, denom, hout);
        long long nt = (long long)N * hcPad;
        gat_bias_act<<<(int)((nt + 255) / 256), 256, 0, stream>>>(hout, biasp, nt, hcPad, relu);
    };

    layer(x,  128, 128, W1lp, W1rp, 64, a1p, b1p, 2, hA, 1);  // 128 -> 2x30 (pad 2x32), relu
    layer(hA,  64,  64, W2lp, W2rp, 32, a2p, b2p, 1, hB, 1);  // 60(padK 64) -> 30 (pad 32), relu
    layer(hB,  32,  32, W3lp, W3rp, 64, a3p, b3p, 2, hA, 0);  // 30(padK 32) -> 2x30 (pad 2x32)
    gemm(hA, 64, Wlp, 16, blin, (float*)d_out, 16, N, 64, 16); // head: 60(padK 64) -> 16
}